// RandomSampleBorderSemanPts_29248727286341
// MI455X (gfx1250) — compile-verified
//
#include <hip/hip_runtime.h>
#include <math.h>

#define B_      10
#define H_      320
#define W_      1024
#define PTS_    500
#define DENSE_  2000
#define N_      (B_ * PTS_)
#define KMEAN_  10

#define PH       15        // by in [-7, 7]
#define PW       23        // bx in [-11, 11]
#define PSTRIDE  24
#define NTHREADS 256
#define NWAVES   8         // wave32: 256/32
#define NOWN     250       // 250 threads x 8 consecutive samples = 2000

#if defined(__has_builtin)
#if __has_builtin(__builtin_amdgcn_global_load_async_to_lds_b32)
#define HAVE_ASYNC_LDS 1
#endif
#if __has_builtin(__builtin_amdgcn_s_wait_asynccnt)
#define HAVE_WAIT_ASYNC 1
#endif
#endif

typedef __attribute__((address_space(1))) int* g32_t;   // global int*
typedef __attribute__((address_space(3))) int* l32_t;   // LDS int*

struct alignas(16) I8 { int v[8]; };   // 2 x global_load_b128

// One-barrier deterministic block reduction (sum,sum).
// buf: 16 floats, caller double-buffers across uses.
__device__ __forceinline__ float2 blockRedSum2(float a, float b, float* buf) {
#pragma unroll
  for (int off = 16; off >= 1; off >>= 1) {
    a += __shfl_xor(a, off, 32);
    b += __shfl_xor(b, off, 32);
  }
  const int wave = threadIdx.x >> 5;
  if ((threadIdx.x & 31) == 0) { buf[wave] = a; buf[NWAVES + wave] = b; }
  __syncthreads();
  float sa = buf[0], sb = buf[NWAVES];
#pragma unroll
  for (int w = 1; w < NWAVES; ++w) { sa += buf[w]; sb += buf[NWAVES + w]; }
  return make_float2(sa, sb);
}

// One-barrier deterministic block reduction (max,min).
__device__ __forceinline__ float2 blockRedMaxMin(float a, float b, float* buf) {
#pragma unroll
  for (int off = 16; off >= 1; off >>= 1) {
    a = fmaxf(a, __shfl_xor(a, off, 32));
    b = fminf(b, __shfl_xor(b, off, 32));
  }
  const int wave = threadIdx.x >> 5;
  if ((threadIdx.x & 31) == 0) { buf[wave] = a; buf[NWAVES + wave] = b; }
  __syncthreads();
  float sa = buf[0], sb = buf[NWAVES];
#pragma unroll
  for (int w = 1; w < NWAVES; ++w) {
    sa = fmaxf(sa, buf[w]); sb = fminf(sb, buf[NWAVES + w]);
  }
  return make_float2(sa, sb);
}

__global__ __launch_bounds__(NTHREADS)
void rsbsp_kernel(const float* __restrict__ disp, const float* __restrict__ fore,
                  const int* __restrict__ cx_, const int* __restrict__ cy_,
                  const int* __restrict__ bx_, const int* __restrict__ by_,
                  float* __restrict__ out) {
  const int row = blockIdx.x;
  const int t   = threadIdx.x;
  const int b   = row / PTS_;
  const int cx  = cx_[row];
  const int cy  = cy_[row];

  __shared__ float patch[PH * PSTRIDE];   // 15x23 disp window (stride 24)
  __shared__ float red[2 * 2 * NWAVES];   // double-buffered reduction scratch
  __shared__ float sob[9];                // |sobel| at 3x3 pool neighbors

  // ---- validity: 3x3 max-pool of |sobel_x(forepred)| at (cy,cx), lanes 0..8 ----
  if (t < 9) {
    const int dy = t / 3 - 1, dx = t % 3 - 1;
    const int y = cy + dy, x = cx + dx;
    const float* fp = fore + (size_t)b * (H_ * W_);
    const float g =
        -fp[(size_t)(y - 1) * W_ + (x - 1)] + fp[(size_t)(y - 1) * W_ + (x + 1)]
        - 2.f * fp[(size_t)y * W_ + (x - 1)] + 2.f * fp[(size_t)y * W_ + (x + 1)]
        - fp[(size_t)(y + 1) * W_ + (x - 1)] + fp[(size_t)(y + 1) * W_ + (x + 1)];
    sob[t] = fabsf(g);
  }

  // ---- stage the 15x23 disp window into LDS via async DMA-to-LDS ----
  const size_t gbase = (size_t)b * (H_ * W_) + (size_t)(cy - 7) * W_ + (size_t)(cx - 11);
  for (int e = t; e < PH * PW; e += NTHREADS) {
    const int py = e / PW, px = e % PW;
#if HAVE_ASYNC_LDS
    __builtin_amdgcn_global_load_async_to_lds_b32(
        (g32_t)(disp + gbase + (size_t)py * W_ + px),
        (l32_t)&patch[py * PSTRIDE + px], 0, 0);
#else
    patch[py * PSTRIDE + px] = disp[gbase + (size_t)py * W_ + px];
#endif
  }
#if HAVE_ASYNC_LDS
#if HAVE_WAIT_ASYNC
  __builtin_amdgcn_s_wait_asynccnt(0);
#else
  asm volatile("s_wait_asynccnt 0" ::: "memory");
#endif
#endif
  __syncthreads();

  // ---- offsets: thread t (<250) owns 8 consecutive samples, 2x b128 per array ----
  const int* bxp = bx_ + (size_t)row * DENSE_;
  const int* byp = by_ + (size_t)row * DENSE_;
  const bool own = (t < NOWN);

  float s[8];
  float vmax = -3.402823466e38f, vmin = 3.402823466e38f, vsum = 0.f;
  if (own) {
    const I8 xs = *(const I8*)(bxp + (size_t)t * 8);
    const I8 ys = *(const I8*)(byp + (size_t)t * 8);
#pragma unroll
    for (int j = 0; j < 8; ++j) {
      const float v = patch[(ys.v[j] + 7) * PSTRIDE + (xs.v[j] + 11)];
      s[j] = v;
      vmax = fmaxf(vmax, v);
      vmin = fminf(vmin, v);
      vsum += v;
    }
  } else {
#pragma unroll
    for (int j = 0; j < 8; ++j) s[j] = 0.f;
  }

  // ---- init means: kL = max, kS = min; total sum computed once.
  //      Buffers alternate (maxmin: buf0, sum: buf1, iter it: buf[it&1]) so a
  //      single barrier per reduction is hazard-free. ----
  const float2 mm = blockRedMaxMin(vmax, vmin, &red[0]);
  float kL = mm.x, kS = mm.y;
  const float2 sa = blockRedSum2(vsum, 0.f, &red[2 * NWAVES]);
  const float sumAll = sa.x;

  // ---- 10 k-means iterations. Since kL >= kS:
  //      |s-kL| <= |s-kS|  <=>  s >= (kL+kS)/2 ; small cluster = complement ----
  float cntL = 0.f;
#pragma unroll 1
  for (int it = 0; it < KMEAN_; ++it) {
    const float mid = 0.5f * (kL + kS);
    float sl = 0.f, cl = 0.f;
    if (own) {
#pragma unroll
      for (int j = 0; j < 8; ++j) {
        if (s[j] >= mid) { sl += s[j]; cl += 1.f; }
      }
    }
    const float2 r = blockRedSum2(sl, cl, &red[(it & 1) * 2 * NWAVES]);
    const float sumL = r.x;
    cntL = r.y;
    kL = sumL / cntL;
    kS = (sumAll - sumL) / ((float)DENSE_ - cntL);
  }

  // ---- final selection + write (thread 0) ----
  if (t == 0) {
    float mx = sob[0];
#pragma unroll
    for (int k = 1; k < 9; ++k) mx = fmaxf(mx, sob[k]);
    const float dC = disp[(size_t)b * (H_ * W_) + (size_t)cy * W_ + cx];
    float mg = (dC > 0.007f) ? mx : 0.f;
    mg = fmaxf(mg, 3.f) - 3.f;
    const bool validRow = mg > 0.1f;
    const bool contrast = ((kL - kS) > 0.005f) && (cntL > 5.f);
    const float keep = (validRow && contrast) ? 1.f : 0.f;
    out[(size_t)row * 2 + 0] = kL * keep;
    out[(size_t)row * 2 + 1] = kS * keep;
  }
}

extern "C" void kernel_launch(void* const* d_in, const int* in_sizes, int n_in,
                              void* d_out, int out_size, void* d_ws, size_t ws_size,
                              hipStream_t stream) {
  (void)in_sizes; (void)n_in; (void)out_size; (void)d_ws; (void)ws_size;
  const float* disp = (const float*)d_in[0];
  const float* fore = (const float*)d_in[1];
  const int*   cx   = (const int*)d_in[2];
  const int*   cy   = (const int*)d_in[3];
  const int*   bx   = (const int*)d_in[4];
  const int*   by   = (const int*)d_in[5];
  float*       out  = (float*)d_out;

  hipLaunchKernelGGL(rsbsp_kernel, dim3(N_), dim3(NTHREADS), 0, stream,
                     disp, fore, cx, cy, bx, by, out);
}